// GraphClassifierMulti_28656021798993
// MI455X (gfx1250) — compile-verified
//
#include <hip/hip_runtime.h>

// Problem constants (from reference)
#define NB 100000          // N nodes
#define EE 500000          // E edges
#define BL 8192            // B links
#define PS 8               // P slots
#define EMBD 64
#define PSD 256
#define DIVD 32
#define RELD 32
#define INPD 64
#define MIDD 288           // PSD + DIVD
#define HDIN 352           // MIDD + EMBD
#define LNKIN 960          // 2*PSD + RELD + MIDD + 2*INPD
#define H1D 128
#define H2D 64

typedef __attribute__((ext_vector_type(2))) float v2f;
typedef __attribute__((ext_vector_type(8))) float v8f;

// CDNA5 fp32 WMMA: D(16x16,f32) = A(16x4,f32) x B(4x16,f32) + C
// A frag (2 VGPR): lanes 0-15 hold row M=lane, K=k0..k0+1; lanes 16-31 K=k0+2..k0+3
// B frag (2 VGPR): lanes 0-15 hold col N=lane, K=k0..k0+1; lanes 16-31 K=k0+2..k0+3
// C/D: VGPR j, lanes 0-15 -> row j, lanes 16-31 -> row j+8; col = lane&15
__device__ __forceinline__ v8f wmma_f32_k4(v2f a, v2f b, v8f c) {
  return __builtin_amdgcn_wmma_f32_16x16x4_f32(
      /*neg_a=*/false, a, /*neg_b=*/false, b,
      /*c_mod=*/(short)0, c, /*reuse_a=*/false, /*reuse_b=*/false);
}

__device__ __forceinline__ float isign_p1(int v) {
  // sign(v+1) as float: v>-1 -> 1, v==-1 -> 0, v<-1 -> -1
  return (v > -1) ? 1.0f : ((v == -1) ? 0.0f : -1.0f);
}

// ---------------------------------------------------------------------------
// Kernel 1: all gathers + on-demand edge_div + mid; fills hd_in and g_rep
// staging buffers; writes head_init / tail_init outputs.
// One block (256 threads, 8 waves) per link b.
// ---------------------------------------------------------------------------
__global__ __launch_bounds__(256) void k_gather_mid(
    const float* __restrict__ node_repr, const float* __restrict__ op_final,
    const float* __restrict__ feat, const int* __restrict__ dst_idx,
    const int* __restrict__ rep_mask, const int* __restrict__ links,
    const int* __restrict__ dist, const int* __restrict__ inter_count,
    const int* __restrict__ edge_ids, const float* __restrict__ W_el,
    const float* __restrict__ b_el, const float* __restrict__ rel_table,
    const float* __restrict__ dist_table, float* __restrict__ hdin,
    float* __restrict__ grep, float* __restrict__ head_init_out,
    float* __restrict__ tail_init_out) {
  __shared__ float weldsT[EMBD * DIVD];   // W_el transposed: [d][o]
  __shared__ float ofb[PS][PSD];          // staged op_final rows per slot
  __shared__ float eaccs[PS][DIVD];       // per-slot signed edge_div

  const int b = blockIdx.x;
  const int t = threadIdx.x;
  const int lane = t & 31;
  const int wv = t >> 5;

  // Preload W_el (32x64) transposed -> lane-indexed inner dim (no LDS bank conflicts)
  for (int i = t; i < DIVD * EMBD; i += 256) {
    int o = i >> 6;
    int d = i & 63;
    weldsT[d * DIVD + o] = W_el[i];
  }
  __syncthreads();

  // ---- edge part: wave wv handles slot p = wv; lane = output channel o ----
  {
    const int p = wv;
    const int eraw = edge_ids[b * PS + p];
    const float es = isign_p1(eraw);
    const int ae = eraw < 0 ? -eraw : eraw;
    const int dst = dst_idx[ae];
    // stage op_final[dst] (256 floats) in LDS (same-wave DS ordering)
    for (int c = 0; c < 8; ++c)
      ofb[p][c * 32 + lane] = op_final[dst * PSD + c * 32 + lane];
    const float n0m = (rep_mask[ae * 4 + 0] == 0) ? 1.f : 0.f;
    const float n1m = (rep_mask[ae * 4 + 1] == 0) ? 1.f : 0.f;
    const float n2m = (rep_mask[ae * 4 + 2] == 0) ? 1.f : 0.f;
    const float n3m = (rep_mask[ae * 4 + 3] == 0) ? 1.f : 0.f;
    const float negsum = n0m + n1m + n2m + n3m;
    const float denom = negsum < 1.f ? 1.f : negsum;
    float f0 = 0.f, f1 = 0.f, f2 = 0.f, f3 = 0.f;
    for (int d = 0; d < EMBD; ++d) {
      const float w = weldsT[d * DIVD + lane];   // conflict-free
      f0 += ofb[p][d] * w;                       // broadcast reads
      f1 += ofb[p][64 + d] * w;
      f2 += ofb[p][128 + d] * w;
      f3 += ofb[p][192 + d] * w;
    }
    const float num = f0 * n0m + f1 * n1m + f2 * n2m + f3 * n3m +
                      negsum * b_el[lane];
    eaccs[p][lane] = es * num / denom;
  }

  // ---- node part: thread t owns component t of the 256-dim sum ----
  float accN = 0.f, cntf = 0.f;
  for (int p = 0; p < PS; ++p) {
    const int iv = inter_count[b * PS + p];       // broadcast load
    if (iv != -1) cntf += 1.f;
    const float ns = isign_p1(iv);
    const int ai = iv < 0 ? -iv : iv;
    accN += ns * node_repr[ai * PSD + t];         // coalesced row gather
  }
  if (cntf < 1.f) cntf = 1.f;
  const float inv = 1.f / cntf;
  const float mnode = accN * inv;
  hdin[b * HDIN + t] = mnode;
  grep[b * LNKIN + 544 + t] = mnode;

  __syncthreads();
  if (t < DIVD) {
    float s = 0.f;
    for (int p = 0; p < PS; ++p) s += eaccs[p][t];
    const float medge = s * inv;
    hdin[b * HDIN + PSD + t] = medge;
    grep[b * LNKIN + 544 + PSD + t] = medge;
  }

  // ---- remaining gathers into g_rep / hd_in / outputs ----
  const int l0 = links[b * 3 + 0];
  const int l1 = links[b * 3 + 1];
  const int l2 = links[b * 3 + 2];
  grep[b * LNKIN + t] = node_repr[l0 * PSD + t];          // head_repr
  grep[b * LNKIN + 256 + t] = node_repr[l2 * PSD + t];    // tail_repr
  if (t < RELD) grep[b * LNKIN + 512 + t] = rel_table[l1 * RELD + t];
  if (t < INPD) {
    const float fh = feat[l0 * INPD + t];
    const float ft = feat[l2 * INPD + t];
    grep[b * LNKIN + 832 + t] = fh;
    grep[b * LNKIN + 896 + t] = ft;
    head_init_out[b * INPD + t] = fh;
    tail_init_out[b * INPD + t] = ft;
    const int dv = dist[b];
    hdin[b * HDIN + MIDD + t] = dist_table[dv * EMBD + t];
  }
}

// ---------------------------------------------------------------------------
// Kernel 2: head_pred / tail_pred = hd_in @ W.T + bias via fp32 WMMA, K=352.
// One wave per 16x16 output tile; 4096 wave-tiles (head+tail).
// ---------------------------------------------------------------------------
__global__ __launch_bounds__(256) void k_pred(
    const float* __restrict__ hdin, const float* __restrict__ Wh,
    const float* __restrict__ bh, const float* __restrict__ Wt,
    const float* __restrict__ bt, float* __restrict__ head_pred,
    float* __restrict__ tail_pred) {
  const int lane = threadIdx.x & 31;
  const int w = blockIdx.x * 8 + (threadIdx.x >> 5);  // 0..4095
  const int which = w >> 11;                          // 0=head, 1=tail
  const int rem = w & 2047;
  const int m0 = (rem >> 2) * 16;
  const int n0 = (rem & 3) * 16;
  const float* W = which ? Wt : Wh;
  const float* bias = which ? bt : bh;
  float* out = which ? tail_pred : head_pred;

  const int half = lane >> 4;
  const int i16 = lane & 15;
  const float* arow = hdin + (m0 + i16) * HDIN + 2 * half;
  const float* brow = W + (n0 + i16) * HDIN + 2 * half;

  v8f acc = {};
  for (int k0 = 0; k0 < HDIN; k0 += 4) {
    v2f a = *(const v2f*)(arow + k0);
    v2f bb = *(const v2f*)(brow + k0);
    acc = wmma_f32_k4(a, bb, acc);
  }
  const float bv = bias[n0 + i16];
#pragma unroll
  for (int j = 0; j < 8; ++j) {
    const int row = m0 + j + 8 * half;
    out[row * EMBD + n0 + i16] = acc[j] + bv;
  }
}

// ---------------------------------------------------------------------------
// Kernel 3: h1 = relu(g_rep @ W1.T + b1) via fp32 WMMA, K=960, NOUT=128.
// ---------------------------------------------------------------------------
__global__ __launch_bounds__(256) void k_mlp1(
    const float* __restrict__ grep, const float* __restrict__ W1,
    const float* __restrict__ b1, float* __restrict__ h1) {
  const int lane = threadIdx.x & 31;
  const int w = blockIdx.x * 8 + (threadIdx.x >> 5);  // 0..4095
  const int m0 = (w >> 3) * 16;
  const int n0 = (w & 7) * 16;

  const int half = lane >> 4;
  const int i16 = lane & 15;
  const float* arow = grep + (m0 + i16) * LNKIN + 2 * half;
  const float* brow = W1 + (n0 + i16) * LNKIN + 2 * half;

  v8f acc = {};
  for (int k0 = 0; k0 < LNKIN; k0 += 4) {
    v2f a = *(const v2f*)(arow + k0);
    v2f bb = *(const v2f*)(brow + k0);
    acc = wmma_f32_k4(a, bb, acc);
  }
  const float bv = b1[n0 + i16];
#pragma unroll
  for (int j = 0; j < 8; ++j) {
    const int row = m0 + j + 8 * half;
    const float v = acc[j] + bv;
    h1[row * H1D + n0 + i16] = v > 0.f ? v : 0.f;
  }
}

// ---------------------------------------------------------------------------
// Kernel 4: h2 = relu(h1 @ W2.T + b2) via WMMA (K=128) into LDS, then
// output = h2 @ W3.T + b3. One block (4 waves) per 16 links.
// ---------------------------------------------------------------------------
__global__ __launch_bounds__(128) void k_mlp2(
    const float* __restrict__ h1, const float* __restrict__ W2,
    const float* __restrict__ b2, const float* __restrict__ W3,
    const float* __restrict__ b3, float* __restrict__ outp) {
  __shared__ float h2s[16 * H2D];
  const int lane = threadIdx.x & 31;
  const int m0 = blockIdx.x * 16;
  const int n0 = (threadIdx.x >> 5) * 16;  // wave -> N tile

  const int half = lane >> 4;
  const int i16 = lane & 15;
  const float* arow = h1 + (m0 + i16) * H1D + 2 * half;
  const float* brow = W2 + (n0 + i16) * H1D + 2 * half;

  v8f acc = {};
  for (int k0 = 0; k0 < H1D; k0 += 4) {
    v2f a = *(const v2f*)(arow + k0);
    v2f bb = *(const v2f*)(brow + k0);
    acc = wmma_f32_k4(a, bb, acc);
  }
  const float bv = b2[n0 + i16];
#pragma unroll
  for (int j = 0; j < 8; ++j) {
    const int row = j + 8 * half;
    const float v = acc[j] + bv;
    h2s[row * H2D + n0 + i16] = v > 0.f ? v : 0.f;
  }
  __syncthreads();
  if (threadIdx.x < 16) {
    float s = b3[0];
    for (int n = 0; n < H2D; ++n) s += h2s[threadIdx.x * H2D + n] * W3[n];
    outp[m0 + threadIdx.x] = s;
  }
}

// ---------------------------------------------------------------------------
extern "C" void kernel_launch(void* const* d_in, const int* in_sizes, int n_in,
                              void* d_out, int out_size, void* d_ws,
                              size_t ws_size, hipStream_t stream) {
  const float* node_repr = (const float*)d_in[0];
  const float* op_final = (const float*)d_in[1];
  const float* feat = (const float*)d_in[2];
  const int* dst_idx = (const int*)d_in[3];
  const int* rep_mask = (const int*)d_in[4];
  const int* links = (const int*)d_in[5];
  const int* dist = (const int*)d_in[6];
  const int* inter = (const int*)d_in[7];
  const int* edge_ids = (const int*)d_in[8];
  const float* W_el = (const float*)d_in[9];
  const float* b_el = (const float*)d_in[10];
  const float* rel_table = (const float*)d_in[11];
  const float* dist_table = (const float*)d_in[12];
  const float* Wh = (const float*)d_in[13];
  const float* bh = (const float*)d_in[14];
  const float* Wt = (const float*)d_in[15];
  const float* bt = (const float*)d_in[16];
  const float* W1 = (const float*)d_in[17];
  const float* b1 = (const float*)d_in[18];
  const float* W2 = (const float*)d_in[19];
  const float* b2 = (const float*)d_in[20];
  const float* W3 = (const float*)d_in[21];
  const float* b3 = (const float*)d_in[22];

  // Output layout: [output(B), head_pred(B*64), tail_pred(B*64),
  //                 head_init(B*64), tail_init(B*64)]
  float* out = (float*)d_out;
  float* out_score = out;
  float* head_pred = out + BL;
  float* tail_pred = head_pred + (size_t)BL * EMBD;
  float* head_init = tail_pred + (size_t)BL * EMBD;
  float* tail_init = head_init + (size_t)BL * INPD;

  // Workspace: hd_in (B*352) | g_rep (B*960) | h1 (B*128)  ~= 47 MB
  float* ws = (float*)d_ws;
  float* hdin = ws;
  float* grep = hdin + (size_t)BL * HDIN;
  float* h1 = grep + (size_t)BL * LNKIN;

  k_gather_mid<<<BL, 256, 0, stream>>>(node_repr, op_final, feat, dst_idx,
                                       rep_mask, links, dist, inter, edge_ids,
                                       W_el, b_el, rel_table, dist_table, hdin,
                                       grep, head_init, tail_init);
  k_pred<<<512, 256, 0, stream>>>(hdin, Wh, bh, Wt, bt, head_pred, tail_pred);
  k_mlp1<<<512, 256, 0, stream>>>(grep, W1, b1, h1);
  k_mlp2<<<512, 128, 0, stream>>>(h1, W2, b2, W3, b3, out_score);
}